// SparseRouter_42623255445546
// MI455X (gfx1250) — compile-verified
//
#include <hip/hip_runtime.h>
#include <math.h>

// Problem constants (match reference)
#define B_TOTAL  32768
#define D_MODEL  1024
#define N_TIERS  3

typedef float v2f __attribute__((ext_vector_type(2)));
typedef float v4f __attribute__((ext_vector_type(4)));
typedef float v8f __attribute__((ext_vector_type(8)));

// ---------------------------------------------------------------------------
// Kernel 1: router logits via V_WMMA_F32_16X16X4_F32 + fused top-2 softmax.
// 256 threads = 8 waves; each wave computes logits for 16 batch rows
// (16x16 output tile, only N=0..2 meaningful), looping K=1024 in steps of 4.
// ---------------------------------------------------------------------------
__global__ __launch_bounds__(256)
void router_gate_kernel(const float* __restrict__ query,
                        const float* __restrict__ W,
                        float* __restrict__ gates_out,
                        float* __restrict__ logits_out)
{
    __shared__ float lds_logits[128 * N_TIERS];   // 8 waves * 16 rows * 3 tiers

    const int lane   = threadIdx.x & 31;
    const int wave   = threadIdx.x >> 5;
    const int n      = lane & 15;     // N column within 16x16 tile
    const int khalf  = lane >> 4;     // 0 -> K={0,1}, 1 -> K={2,3}
    const int rowBase = blockIdx.x * 128 + wave * 16;

    // A (16x4 f32): lane(n,khalf) supplies query[rowBase+n][k + 2*khalf + {0,1}]
    const float* qptr = query + (size_t)(rowBase + n) * D_MODEL + 2 * khalf;
    // B (4x16 f32): lane n supplies W[n][k + 2*khalf + {0,1}] (0 for n>=3)
    const float  wmask = (n < N_TIERS) ? 1.0f : 0.0f;
    const float* wptr  = W + (size_t)((n < N_TIERS) ? n : 0) * D_MODEL + 2 * khalf;

    v8f c = {};
#pragma unroll 4
    for (int kk = 0; kk < D_MODEL; kk += 4) {
        v2f a  = *(const v2f*)(qptr + kk);
        v2f bw = *(const v2f*)(wptr + kk);
        v2f b;
        b.x = bw.x * wmask;
        b.y = bw.y * wmask;
        // D = A*B + C : v_wmma_f32_16x16x4_f32
        c = __builtin_amdgcn_wmma_f32_16x16x4_f32(
                /*neg_a=*/false, a, /*neg_b=*/false, b,
                /*c_mod=*/(short)0, c, /*reuse_a=*/false, /*reuse_b=*/false);
    }

    // C/D layout: VGPR r -> M=r (lanes 0-15), M=r+8 (lanes 16-31); N = lane&15.
    if (n < N_TIERS) {
#pragma unroll
        for (int r = 0; r < 8; ++r) {
            int m = r + 8 * khalf;  // row within this wave's 16-row tile
            lds_logits[(wave * 16 + m) * N_TIERS + n] = c[r];
        }
    }
    __syncthreads();

    // Lanes 0-15 of each wave: one batch row each -> top-2 softmax gating.
    if (lane < 16) {
        const int lrow = wave * 16 + lane;
        const int grow = blockIdx.x * 128 + lrow;
        const float l0 = lds_logits[lrow * N_TIERS + 0];
        const float l1 = lds_logits[lrow * N_TIERS + 1];
        const float l2 = lds_logits[lrow * N_TIERS + 2];

        // argmax, lowest index wins ties (matches lax.top_k)
        int   i0 = 0; float v0 = l0;
        if (l1 > v0) { v0 = l1; i0 = 1; }
        if (l2 > v0) { v0 = l2; i0 = 2; }
        // second best among the remaining two (lower index wins ties)
        const int   ia = (i0 == 0) ? 1 : 0;
        const int   ib = (i0 == 2) ? 1 : 2;
        const float va = (ia == 0) ? l0 : l1;
        const float vb = (ib == 1) ? l1 : l2;
        int   i1 = ia; float v1 = va;
        if (vb > v1) { v1 = vb; i1 = ib; }

        // softmax over the two kept logits (v0 >= v1 -> numerically stable)
        const float e1 = __expf(v1 - v0);
        const float w0 = 1.0f / (1.0f + e1);
        const float w1 = e1 * w0;

        const float g0 = (i0 == 0) ? w0 : ((i1 == 0) ? w1 : 0.0f);
        const float g1 = (i0 == 1) ? w0 : ((i1 == 1) ? w1 : 0.0f);
        const float g2 = (i0 == 2) ? w0 : ((i1 == 2) ? w1 : 0.0f);

        gates_out[grow * N_TIERS + 0] = g0;
        gates_out[grow * N_TIERS + 1] = g1;
        gates_out[grow * N_TIERS + 2] = g2;
        logits_out[grow * N_TIERS + 0] = l0;
        logits_out[grow * N_TIERS + 1] = l1;
        logits_out[grow * N_TIERS + 2] = l2;
    }
}

// ---------------------------------------------------------------------------
// Kernel 2: merged[b,:] = sum_t gates[b,t] * tier_outputs[b,t,:]
// Pure HBM-streaming kernel: one float4 per thread, three b128 loads,
// one b128 store; gates broadcast (same address across a row's threads).
// ---------------------------------------------------------------------------
__global__ __launch_bounds__(256)
void merge_kernel(const float* __restrict__ tier,
                  const float* __restrict__ gates,
                  float* __restrict__ merged)
{
    const int gid = blockIdx.x * blockDim.x + threadIdx.x;
    const int b   = gid >> 8;          // D_MODEL/4 = 256 float4 chunks per row
    const int dq  = gid & 255;

    const float g0 = gates[b * N_TIERS + 0];
    const float g1 = gates[b * N_TIERS + 1];
    const float g2 = gates[b * N_TIERS + 2];

    const float* base = tier + (size_t)b * N_TIERS * D_MODEL + dq * 4;
    const v4f t0 = *(const v4f*)(base);
    const v4f t1 = *(const v4f*)(base + D_MODEL);
    const v4f t2 = *(const v4f*)(base + 2 * D_MODEL);

    v4f out = t0 * g0 + t1 * g1 + t2 * g2;
    *(v4f*)(merged + (size_t)b * D_MODEL + dq * 4) = out;
}

// ---------------------------------------------------------------------------
// d_in[0] = tier_outputs (B, 3, D) f32
// d_in[1] = query        (B, D)    f32
// d_in[2] = W            (3, D)    f32
// d_out   = [ merged (B*D) | gate_weights (B*3) | router_logits (B*3) ] f32
// ---------------------------------------------------------------------------
extern "C" void kernel_launch(void* const* d_in, const int* in_sizes, int n_in,
                              void* d_out, int out_size, void* d_ws, size_t ws_size,
                              hipStream_t stream) {
    const float* tier  = (const float*)d_in[0];
    const float* query = (const float*)d_in[1];
    const float* W     = (const float*)d_in[2];

    float* out       = (float*)d_out;
    float* merged    = out;
    float* gates_out = out + (size_t)B_TOTAL * D_MODEL;
    float* logits_out = gates_out + (size_t)B_TOTAL * N_TIERS;

    // Kernel 1: 128 rows per block (8 waves x 16 rows) -> 256 blocks
    router_gate_kernel<<<B_TOTAL / 128, 256, 0, stream>>>(query, W, gates_out, logits_out);

    // Kernel 2: B*D/4 float4 chunks, 256 threads/block -> 32768 blocks
    merge_kernel<<<(B_TOTAL * (D_MODEL / 4)) / 256, 256, 0, stream>>>(tier, gates_out, merged);
}